// GATModel_59004260713103
// MI455X (gfx1250) — compile-verified
//
#include <hip/hip_runtime.h>

typedef __attribute__((ext_vector_type(2))) float v2f;
typedef __attribute__((ext_vector_type(8))) float v8f;

#define NNODES 50000
#define NEDGES 800000
#define ETOT   850000      // E + N self loops
#define FIN    128
#define HIDC   64
#define NHEADS 4
#define EPSBN  1e-5f

// =====================================================================
// fp32 WMMA GEMM: C[M,Nc] = A[M,K] @ B[K,Nc], all row-major. K templated
// (128 / 256) so the K-loop has a compile-time trip count.
//
// Block = 256 threads (8 waves). The block owns one 16-column tile:
//   * all 8 waves cooperatively stage the K x 16 B panel into LDS once
//   * each wave then computes 5 stacked 16x16 M-tiles (80 rows), reusing
//     the LDS-resident B fragment across the 5 WMMAs of each K-step.
// ISA layout (CDNA5 ISA):
//   A 16x4 : lanes 0-15 hold K={0,1} (v2f), lanes 16-31 hold K={2,3}
//   B 4x16 : symmetric; C/D: v8f, rows 0-7 lanes 0-15 / rows 8-15 lanes 16-31
// LDS fragment reads are bank-conflict free (rows k0 and k0+2 are 32
// floats apart -> 32 distinct banks across the wave).
// =====================================================================
template <int K>
__global__ __launch_bounds__(256) void gemm_wmma_f32(
    const float* __restrict__ A, const float* __restrict__ B,
    float* __restrict__ C, int Nc, int n_wave_jobs)
{
  __shared__ float Bs[K * 16];

  const int tid  = threadIdx.x;
  const int lane = tid & 31;
  const int wave = tid >> 5;
  const int half = lane >> 4;          // which K-pair of the k0..k0+3 step
  const int ml   = lane & 15;
  const int col0 = blockIdx.x * 16;

  // cooperative stage of the B panel (K x 16) into LDS, coalesced
  #pragma unroll
  for (int i = tid; i < K * 16; i += 256) {
    const int r = i >> 4, cc = i & 15;
    Bs[i] = B[(size_t)r * Nc + col0 + cc];
  }
  __syncthreads();

  const int job = blockIdx.y * 8 + wave;    // one job = 5 M-tiles = 80 rows
  if (job < n_wave_jobs) {                  // wave-uniform: EXEC stays full
    const int row_base = job * 80;

    v8f acc[5];
    #pragma unroll
    for (int t = 0; t < 5; ++t)
      acc[t] = (v8f){0.f,0.f,0.f,0.f,0.f,0.f,0.f,0.f};

    #pragma unroll 4
    for (int k0 = 0; k0 < K; k0 += 4) {
      v2f b;
      b.x = Bs[(k0 + 2*half + 0) * 16 + ml];
      b.y = Bs[(k0 + 2*half + 1) * 16 + ml];
      const float* ap = A + (size_t)(row_base + ml) * K + (k0 + 2*half);
      #pragma unroll
      for (int t = 0; t < 5; ++t) {
        v2f a = *(const v2f*)(ap + (size_t)(16 * t) * K);
        // 8 args: (neg_a, A, neg_b, B, c_mod, C, reuse_a, reuse_b)
        acc[t] = __builtin_amdgcn_wmma_f32_16x16x4_f32(
            false, a, false, b, (short)0, acc[t], false, false);
      }
    }

    #pragma unroll
    for (int t = 0; t < 5; ++t) {
      float* cp = C + (size_t)(row_base + 16*t + 8*half) * Nc + col0 + ml;
      #pragma unroll
      for (int i = 0; i < 8; ++i)
        cp[(size_t)i * Nc] = acc[t][i];
    }
  }
}

// ---------------- per-node attention scores ----------------
__global__ void att_scores(const float* __restrict__ hf,
                           const float* __restrict__ a_s,
                           const float* __restrict__ a_d,
                           float* __restrict__ s_src,
                           float* __restrict__ s_dst,
                           int n, int heads, int c)
{
  int idx = blockIdx.x * blockDim.x + threadIdx.x;
  if (idx >= n * heads) return;
  int node = idx / heads, h = idx % heads;
  const float* f  = hf  + (size_t)node * heads * c + (size_t)h * c;
  const float* as = a_s + (size_t)h * c;
  const float* ad = a_d + (size_t)h * c;
  float ss = 0.f, sd = 0.f;
  #pragma unroll 4
  for (int i = 0; i < c; ++i) { float v = f[i]; ss += v * as[i]; sd += v * ad[i]; }
  s_src[idx] = ss; s_dst[idx] = sd;
}

// monotonic uint encoding for float atomicMax (exact segment-max)
__device__ __forceinline__ unsigned f2key(float f) {
  unsigned u = __float_as_uint(f);
  return (u & 0x80000000u) ? ~u : (u | 0x80000000u);
}
__device__ __forceinline__ float key2f(unsigned k) {
  return __uint_as_float((k & 0x80000000u) ? (k ^ 0x80000000u) : ~k);
}

__device__ __forceinline__ void edge_src_dst(const int* __restrict__ ei, int e,
                                             int& src, int& dst) {
  if (e < NEDGES) { src = ei[e]; dst = ei[NEDGES + e]; }
  else            { src = dst = e - NEDGES; }          // self loop
}

// pass 1: leaky_relu logits + segment max (encoded atomicMax)
__global__ void edge_logits(const int* __restrict__ ei,
                            const float* __restrict__ s_src,
                            const float* __restrict__ s_dst,
                            float* __restrict__ e_buf,
                            unsigned* __restrict__ emax_key, int heads)
{
  int idx = blockIdx.x * blockDim.x + threadIdx.x;
  if (idx >= ETOT * heads) return;
  int e = idx / heads, h = idx % heads;
  int src, dst; edge_src_dst(ei, e, src, dst);
  float v = s_src[src * heads + h] + s_dst[dst * heads + h];
  v = v > 0.f ? v : 0.2f * v;          // leaky_relu(0.2)
  e_buf[idx] = v;
  atomicMax(&emax_key[dst * heads + h], f2key(v));
}

// pass 2: exp(e - max) + segment sum for denominator
__global__ void edge_exp(const int* __restrict__ ei,
                         float* __restrict__ e_buf,
                         const unsigned* __restrict__ emax_key,
                         float* __restrict__ denom, int heads)
{
  int idx = blockIdx.x * blockDim.x + threadIdx.x;
  if (idx >= ETOT * heads) return;
  int e = idx / heads, h = idx % heads;
  int src, dst; edge_src_dst(ei, e, src, dst);
  float m  = key2f(emax_key[dst * heads + h]);
  float ee = expf(e_buf[idx] - m);
  e_buf[idx] = ee;
  atomicAdd(&denom[dst * heads + h], ee);
}

// pass 3: alpha-weighted scatter-sum, vectorized: one b128 load + 4 L2
// atomics per thread. Lanes within a wave cover consecutive channel
// quads of the same edge -> coalesced hf[src] reads and agg[dst] atomics.
__global__ void edge_aggregate(const int* __restrict__ ei,
                               const float* __restrict__ e_buf,
                               const float* __restrict__ denom,
                               const float* __restrict__ hf,
                               float* __restrict__ agg, int heads, int c)
{
  const int hc_tot = heads * c;
  const int hq     = hc_tot >> 2;                 // channel quads per edge
  long long tid   = (long long)blockIdx.x * blockDim.x + threadIdx.x;
  long long total = (long long)ETOT * hq;
  if (tid >= total) return;
  const int e  = (int)(tid / hq);
  const int q  = (int)(tid % hq);
  const int hc = q << 2;
  const int h  = hc / c;
  int src, dst; edge_src_dst(ei, e, src, dst);
  const float alpha = e_buf[(size_t)e * heads + h] /
                      (denom[dst * heads + h] + 1e-16f);
  const float4 v = *(const float4*)(hf + (size_t)src * hc_tot + hc);
  float* dp = agg + (size_t)dst * hc_tot + hc;
  atomicAdd(dp + 0, v.x * alpha);
  atomicAdd(dp + 1, v.y * alpha);
  atomicAdd(dp + 2, v.z * alpha);
  atomicAdd(dp + 3, v.w * alpha);
}

// bias + BatchNorm (eval) + optional ELU; safe in-place
__global__ void post_bn(const float* __restrict__ agg,
                        const float* __restrict__ bias,
                        const float* __restrict__ g,
                        const float* __restrict__ bt,
                        const float* __restrict__ m,
                        const float* __restrict__ v,
                        float* __restrict__ out, int n, int ch, int do_elu)
{
  long long idx = (long long)blockIdx.x * blockDim.x + threadIdx.x;
  if (idx >= (long long)n * ch) return;
  int cix = (int)(idx % ch);
  float x = agg[idx] + bias[cix];
  x = (x - m[cix]) * rsqrtf(v[cix] + EPSBN) * g[cix] + bt[cix];
  if (do_elu) x = x > 0.f ? x : expm1f(x);
  out[idx] = x;
}

// final 64 -> 2 linear (tiny N dim; plain VALU dot)
__global__ void final_linear(const float* __restrict__ h,
                             const float* __restrict__ Wc,
                             const float* __restrict__ bc,
                             float* __restrict__ out)
{
  int n = blockIdx.x * blockDim.x + threadIdx.x;
  if (n >= NNODES) return;
  float o0 = bc[0], o1 = bc[1];
  const float* r = h + (size_t)n * HIDC;
  #pragma unroll 8
  for (int i = 0; i < HIDC; ++i) {
    float x = r[i];
    o0 += x * Wc[2*i + 0];
    o1 += x * Wc[2*i + 1];
  }
  out[2*n + 0] = o0;
  out[2*n + 1] = o1;
}

__global__ void fill_u32(unsigned* __restrict__ p, unsigned val, long long n)
{
  long long i = (long long)blockIdx.x * blockDim.x + threadIdx.x;
  if (i < n) p[i] = val;
}

static inline unsigned gblk(long long total, int bs) {
  return (unsigned)((total + bs - 1) / bs);
}

extern "C" void kernel_launch(void* const* d_in, const int* in_sizes, int n_in,
                              void* d_out, int out_size, void* d_ws, size_t ws_size,
                              hipStream_t stream)
{
  const float* x  = (const float*)d_in[0];
  const int*   ei = (const int*)  d_in[1];
  const float* Wc = (const float*)d_in[26];
  const float* bc = (const float*)d_in[27];

  // workspace carve-up (~120 MB total)
  char* ws = (char*)d_ws;
  float*    hf    = (float*)ws;    ws += (size_t)NNODES * 256 * sizeof(float);
  float*    agg   = (float*)ws;    ws += (size_t)NNODES * 256 * sizeof(float);
  float*    ssrc  = (float*)ws;    ws += (size_t)NNODES * NHEADS * sizeof(float);
  float*    sdst  = (float*)ws;    ws += (size_t)NNODES * NHEADS * sizeof(float);
  unsigned* emax  = (unsigned*)ws; ws += (size_t)NNODES * NHEADS * sizeof(unsigned);
  float*    denom = (float*)ws;    ws += (size_t)NNODES * NHEADS * sizeof(float);
  float*    ebuf  = (float*)ws;    ws += (size_t)ETOT   * NHEADS * sizeof(float);

  const int n_wave_jobs = (NNODES / 16) / 5;     // 625 jobs of 80 rows

  const float* in = x;
  for (int l = 0; l < 3; ++l) {
    int base = 2 + l * 8;
    const float* W  = (const float*)d_in[base + 0];
    const float* as = (const float*)d_in[base + 1];
    const float* ad = (const float*)d_in[base + 2];
    const float* bb = (const float*)d_in[base + 3];
    const float* g  = (const float*)d_in[base + 4];
    const float* bt = (const float*)d_in[base + 5];
    const float* m  = (const float*)d_in[base + 6];
    const float* vv = (const float*)d_in[base + 7];

    const int heads = (l < 2) ? NHEADS : 1;
    const int c     = HIDC;
    const int hc    = heads * c;                 // 256, 256, 64

    // GEMM: hf = in @ W   (8 waves/block, 80 rows per wave-job)
    dim3 ggrid(hc / 16, (unsigned)((n_wave_jobs + 7) / 8));
    if (l == 0)
      gemm_wmma_f32<FIN><<<ggrid, 256, 0, stream>>>(in, W, hf, hc, n_wave_jobs);
    else
      gemm_wmma_f32<NHEADS * HIDC><<<ggrid, 256, 0, stream>>>(in, W, hf, hc, n_wave_jobs);

    const int nh = NNODES * heads;
    att_scores<<<gblk(nh, 256), 256, 0, stream>>>(hf, as, ad, ssrc, sdst,
                                                  NNODES, heads, c);

    fill_u32<<<gblk((long long)NNODES * hc, 256), 256, 0, stream>>>(
        (unsigned*)agg, 0u, (long long)NNODES * hc);
    fill_u32<<<gblk(nh, 256), 256, 0, stream>>>(emax, 0u, nh);
    fill_u32<<<gblk(nh, 256), 256, 0, stream>>>((unsigned*)denom, 0u, nh);

    const long long eh = (long long)ETOT * heads;
    edge_logits<<<gblk(eh, 256), 256, 0, stream>>>(ei, ssrc, sdst, ebuf, emax, heads);
    edge_exp   <<<gblk(eh, 256), 256, 0, stream>>>(ei, ebuf, emax, denom, heads);

    const long long tot4 = (long long)ETOT * (hc >> 2);
    edge_aggregate<<<gblk(tot4, 256), 256, 0, stream>>>(ei, ebuf, denom, hf, agg,
                                                        heads, c);

    post_bn<<<gblk((long long)NNODES * hc, 256), 256, 0, stream>>>(
        agg, bb, g, bt, m, vv, agg, NNODES, hc, (l < 2) ? 1 : 0);

    in = agg;   // in-place BN output becomes next layer's input
  }

  final_linear<<<gblk(NNODES, 256), 256, 0, stream>>>(in, Wc, bc, (float*)d_out);
}